// Attn_9036611191263
// MI455X (gfx1250) — compile-verified
//
#include <hip/hip_runtime.h>
#include <stdint.h>

// Attention energies + softmax for MI455X (gfx1250).
// Memory-bound (0.5 flop/byte over 256MB) -> stream E once via async LDS copies,
// do the dot products with V_WMMA_F32_16X16X4_F32 (B operand = broadcast H chunk).

#define SEQ_LEN 2048
#define BATCH   32
#define HIDDEN  1024

typedef float v2f __attribute__((ext_vector_type(2)));
typedef float v4f __attribute__((ext_vector_type(4)));
typedef float v8f __attribute__((ext_vector_type(8)));
typedef int   i32x4 __attribute__((ext_vector_type(4)));

#if defined(__gfx1250__) && __has_builtin(__builtin_amdgcn_global_load_async_to_lds_b128)
#define USE_ASYNC_LDS 1
#else
#define USE_ASYNC_LDS 0
#endif

#if USE_ASYNC_LDS
// Builtin signature (from hipcc diagnostic): param0 is `int4 addrspace(1)*`.
typedef __attribute__((address_space(1))) i32x4 g_i32x4;
typedef __attribute__((address_space(3))) i32x4 l_i32x4;
#define ASYNC_B128(gp, lp) \
    __builtin_amdgcn_global_load_async_to_lds_b128((g_i32x4*)(gp), (l_i32x4*)(lp), 0, 0)
#endif

__device__ __forceinline__ void async_wait_0() {
#if USE_ASYNC_LDS
#if __has_builtin(__builtin_amdgcn_s_wait_asynccnt)
    __builtin_amdgcn_s_wait_asynccnt(0);
#else
    asm volatile("s_wait_asynccnt 0" ::: "memory");
#endif
#endif
}

__device__ __forceinline__ void async_wait_8() {
#if USE_ASYNC_LDS
#if __has_builtin(__builtin_amdgcn_s_wait_asynccnt)
    __builtin_amdgcn_s_wait_asynccnt(8);
#else
    asm volatile("s_wait_asynccnt 8" ::: "memory");
#endif
#endif
}

// ---------------------------------------------------------------------------
// Kernel 1: energies[b][s] = dot(E[s,b,:], H[b,:])
// One wave (32 threads) computes a 16-row s-tile for one batch b.
// ---------------------------------------------------------------------------
__global__ __launch_bounds__(32)
void energies_wmma_kernel(const float* __restrict__ enc,   // [S, B, H]
                          const float* __restrict__ hid,   // [B, H]
                          float* __restrict__ energies)    // [B, S] (transposed!)
{
    constexpr int CH     = 64;            // h-chunk per LDS tile
    constexpr int NCHUNK = HIDDEN / CH;   // 16
    constexpr int ROWP   = CH + 4;        // 68 floats: 272B rows -> 16B aligned,
                                          // bank offset 4/row -> conflict-free

    __shared__ float tile[2][16][ROWP];   // double-buffered E tile
    __shared__ float hrow[HIDDEN];        // full H[b,:] row (4KB)

    const int lane  = threadIdx.x;        // 0..31 (wave32)
    const int s0    = blockIdx.x * 16;
    const int b     = blockIdx.y;

    const int m     = lane & 15;          // A-matrix row this lane feeds
    const int kbase = (lane >> 4) << 1;   // 0 (lanes 0-15) or 2 (lanes 16-31)

    // ---- stage H[b,:] into LDS (8 x b128 per wave) ----
    {
        const float* hsrc = hid + (size_t)b * HIDDEN;
        #pragma unroll
        for (int j = 0; j < 8; ++j) {
            const int f = j * 128 + lane * 4;
#if USE_ASYNC_LDS
            ASYNC_B128(hsrc + f, &hrow[f]);
#else
            *(v4f*)&hrow[f] = *(const v4f*)(hsrc + f);
#endif
        }
    }

    // ---- chunk loader: E[s0..s0+15, b, c*64 .. c*64+63] -> tile[buf] ----
    const float* gbase0 = enc + ((size_t)s0 * BATCH + b) * HIDDEN;
    auto load_chunk = [&](int c, int buf) {
        const float* gbase = gbase0 + c * CH;
        #pragma unroll
        for (int j = 0; j < 8; ++j) {
            const int f    = j * 128 + lane * 4;  // flat float index in 16x64 tile
            const int srow = f >> 6;              // 0..15
            const int hcol = f & 63;              // 0..60 step 4
            const float* g = gbase + (size_t)srow * (BATCH * HIDDEN) + hcol;
#if USE_ASYNC_LDS
            ASYNC_B128(g, &tile[buf][srow][hcol]);
#else
            *(v4f*)&tile[buf][srow][hcol] = *(const v4f*)g;
#endif
        }
    };

    load_chunk(0, 0);
    async_wait_0();               // H row + chunk 0 resident

    v8f acc = {0.f, 0.f, 0.f, 0.f, 0.f, 0.f, 0.f, 0.f};

    for (int c = 0; c < NCHUNK; ++c) {
        const int buf = c & 1;
        const bool more = (c + 1) < NCHUNK;
        if (more) load_chunk(c + 1, buf ^ 1);
        if (c > 0) {
            if (more) async_wait_8();   // chunk c done; <=8 in flight for c+1
            else      async_wait_0();   // last chunk: drain
        }

        const float* hbase = &hrow[c * CH];
        #pragma unroll
        for (int k = 0; k < CH; k += 4) {
            // A: 16x4 f32 tile rows (lane m, K-pair kbase)
            v2f a  = *(const v2f*)&tile[buf][m][k + kbase];
            // B: 4x16 with every column = H[b, c*64+k .. +3] (broadcast)
            v2f bb = *(const v2f*)&hbase[k + kbase];
            acc = __builtin_amdgcn_wmma_f32_16x16x4_f32(
                /*neg_a=*/false, a, /*neg_b=*/false, bb,
                /*c_mod=*/(short)0, acc, /*reuse_a=*/false, /*reuse_b=*/false);
        }
    }

    // D layout: vgpr r, lanes 0-15 -> M=r, lanes 16-31 -> M=r+8; all N identical.
    float* erow = energies + (size_t)b * SEQ_LEN + s0;
    if (lane == 0) {
        #pragma unroll
        for (int r = 0; r < 8; ++r) erow[r] = acc[r];
    } else if (lane == 16) {
        #pragma unroll
        for (int r = 0; r < 8; ++r) erow[8 + r] = acc[r];
    }
}

// ---------------------------------------------------------------------------
// Kernel 2: softmax over s for each b; out[b][s] (== attn.T[:,None,:] flat)
// ---------------------------------------------------------------------------
__global__ __launch_bounds__(256)
void softmax_seq_kernel(const float* __restrict__ energies,  // [B, S]
                        float* __restrict__ out)             // [B, 1, S]
{
    __shared__ float red[256];
    const int b = blockIdx.x;
    const int t = threadIdx.x;
    const float* e = energies + (size_t)b * SEQ_LEN;

    float v[8];
    float mx = -__builtin_inff();
    #pragma unroll
    for (int i = 0; i < 8; ++i) {
        v[i] = e[i * 256 + t];
        mx = fmaxf(mx, v[i]);
    }
    red[t] = mx;
    __syncthreads();
    for (int off = 128; off > 0; off >>= 1) {
        if (t < off) red[t] = fmaxf(red[t], red[t + off]);
        __syncthreads();
    }
    mx = red[0];
    __syncthreads();

    float s = 0.f;
    #pragma unroll
    for (int i = 0; i < 8; ++i) {
        v[i] = __expf(v[i] - mx);
        s += v[i];
    }
    red[t] = s;
    __syncthreads();
    for (int off = 128; off > 0; off >>= 1) {
        if (t < off) red[t] += red[t + off];
        __syncthreads();
    }
    const float inv = 1.0f / red[0];

    float* o = out + (size_t)b * SEQ_LEN;
    #pragma unroll
    for (int i = 0; i < 8; ++i) o[i * 256 + t] = v[i] * inv;
}

// ---------------------------------------------------------------------------
extern "C" void kernel_launch(void* const* d_in, const int* in_sizes, int n_in,
                              void* d_out, int out_size, void* d_ws, size_t ws_size,
                              hipStream_t stream) {
    (void)in_sizes; (void)n_in; (void)out_size; (void)ws_size;
    const float* hid = (const float*)d_in[0];   // [BATCH, HIDDEN]
    const float* enc = (const float*)d_in[1];   // [SEQ_LEN, BATCH, HIDDEN]
    float* out       = (float*)d_out;           // [BATCH, 1, SEQ_LEN]
    float* energies  = (float*)d_ws;            // [BATCH, SEQ_LEN] scratch (256KB)

    dim3 g1(SEQ_LEN / 16, BATCH);
    energies_wmma_kernel<<<g1, 32, 0, stream>>>(enc, hid, energies);
    softmax_seq_kernel<<<BATCH, 256, 0, stream>>>(energies, out);
}